// CoTFlow_18760417149533
// MI455X (gfx1250) — compile-verified
//
#include <hip/hip_runtime.h>
#include <hip/hip_bf16.h>

typedef float v2f __attribute__((ext_vector_type(2)));
typedef float v8f __attribute__((ext_vector_type(8)));

#define D_INN   12
#define HALFD   6
#define HID     192
#define NLAYERS 24
#define NBASES  66
#define SYMD    12
#define OUTD    6
#define BATCH   16384
#define EPSP    1e-3f
#define CLAMPC  2.0f

// ---------------------------------------------------------------------------
// WMMA f32 16x16x4 helpers (wave32). Layouts per CDNA5 ISA 7.12.2:
//  A (16x4 f32): lanes 0-15 -> row M=lane, K={k0,k0+1}; lanes 16-31 -> K={k0+2,k0+3}
//  B (4x16 f32): lanes 0-15 -> col N=lane, K={k0,k0+1}; lanes 16-31 -> K={k0+2,k0+3}
//  C/D (16x16):  VGPR r -> M=r (lanes 0-15) / M=r+8 (lanes 16-31), N=lane%16
// All guarded fragment builders are branch-free: clamp address, select value.
// ---------------------------------------------------------------------------
__device__ __forceinline__ v8f wmma4(v2f a, v2f b, v8f c) {
  return __builtin_amdgcn_wmma_f32_16x16x4_f32(false, a, false, b, (short)0, c,
                                               false, false);
}

// A fragment from LDS (16 rows, row stride `stride`), K-guarded (branch-free).
__device__ __forceinline__ v2f aT(const float* lds, int stride, int k0, int kvalid) {
  int l = threadIdx.x & 31;
  int m = l & 15;
  int kk = k0 + ((l >> 4) << 1);
  bool ok0 = kk < kvalid, ok1 = (kk + 1) < kvalid;
  float x = lds[m * stride + (ok0 ? kk : 0)];
  float y = lds[m * stride + (ok1 ? kk + 1 : 0)];
  v2f a;
  a.x = ok0 ? x : 0.f;
  a.y = ok1 ? y : 0.f;
  return a;
}
// Fast A fragment (K known in-range).
__device__ __forceinline__ v2f aT_f(const float* lds, int stride, int k0) {
  int l = threadIdx.x & 31;
  int m = l & 15;
  int kk = k0 + ((l >> 4) << 1);
  v2f a;
  a.x = lds[m * stride + kk];
  a.y = lds[m * stride + kk + 1];
  return a;
}

// B fragment for X @ W^T, W (Ntot x Ktot) row-major: B[k][n] = W[nb+n][k].
// K-guarded, branch-free (two clamped b32 loads + selects). n assumed valid.
__device__ __forceinline__ v2f bT_k(const float* W, int ldw, int nbase, int k0,
                                    int kvalid) {
  int l = threadIdx.x & 31;
  int n = l & 15;
  int kk = k0 + ((l >> 4) << 1);
  bool ok0 = kk < kvalid, ok1 = (kk + 1) < kvalid;
  const float* row = W + (size_t)(nbase + n) * ldw;
  float x = row[ok0 ? kk : 0];
  float y = row[ok1 ? kk + 1 : 0];
  v2f b;
  b.x = ok0 ? x : 0.f;
  b.y = ok1 ? y : 0.f;
  return b;
}
// N-guarded only (K in-range): clamp row, zero-select result.
__device__ __forceinline__ v2f bT_n(const float* W, int ldw, int nbase, int k0,
                                    int nvalid) {
  int l = threadIdx.x & 31;
  int n = l & 15;
  int kk = k0 + ((l >> 4) << 1);
  bool nok = n < nvalid;
  const float* p = W + (size_t)(nbase + (nok ? n : 0)) * ldw + kk;
  float x = p[0], y = p[1];
  v2f b;
  b.x = nok ? x : 0.f;
  b.y = nok ? y : 0.f;
  return b;
}
// Fast (all in-range): one contiguous float2.
__device__ __forceinline__ v2f bT_f(const float* W, int ldw, int nbase, int k0) {
  int l = threadIdx.x & 31;
  int n = l & 15;
  int kk = k0 + ((l >> 4) << 1);
  const float* p = W + (size_t)(nbase + n) * ldw + kk;
  v2f b; b.x = p[0]; b.y = p[1];
  return b;
}

// B fragment for X @ W, W (Ktot x Ntot) row-major: B[k][n] = W[k][nb+n].
// N-guarded only, branch-free.
__device__ __forceinline__ v2f bD_n(const float* W, int ldw, int nbase, int k0,
                                    int nvalid) {
  int l = threadIdx.x & 31;
  int n = l & 15;
  int kk = k0 + ((l >> 4) << 1);
  bool nok = n < nvalid;
  int nc = nok ? n : 0;
  float x = W[(size_t)kk * ldw + nbase + nc];
  float y = W[(size_t)(kk + 1) * ldw + nbase + nc];
  v2f b;
  b.x = nok ? x : 0.f;
  b.y = nok ? y : 0.f;
  return b;
}
__device__ __forceinline__ v2f bD_f(const float* W, int ldw, int nbase, int k0) {
  int l = threadIdx.x & 31;
  int n = l & 15;
  int kk = k0 + ((l >> 4) << 1);
  v2f b;
  b.x = W[(size_t)kk * ldw + nbase + n];
  b.y = W[(size_t)(kk + 1) * ldw + nbase + n];
  return b;
}

__device__ __forceinline__ void store_tile(float* lds, int stride, int nbase, v8f acc) {
  int l = threadIdx.x & 31;
  int col = l & 15;
  int mo = (l < 16) ? 0 : 8;
#pragma unroll
  for (int r = 0; r < 8; ++r) lds[(r + mo) * stride + nbase + col] = acc[r];
}
__device__ __forceinline__ void store_relu(float* lds, int stride, int nbase, v8f acc) {
  int l = threadIdx.x & 31;
  int col = l & 15;
  int mo = (l < 16) ? 0 : 8;
#pragma unroll
  for (int r = 0; r < 8; ++r)
    lds[(r + mo) * stride + nbase + col] = fmaxf(acc[r], 0.f);
}
// Store acc * 1{mask>0}; mask from LDS matrix with same (m,col) indexing.
__device__ __forceinline__ void store_mask(float* lds, int stride, int nbase, v8f acc,
                                           const float* mlds, int mstride) {
  int l = threadIdx.x & 31;
  int col = l & 15;
  int mo = (l < 16) ? 0 : 8;
#pragma unroll
  for (int r = 0; r < 8; ++r) {
    float m = (mlds[(r + mo) * mstride + nbase + col] > 0.f) ? 1.f : 0.f;
    lds[(r + mo) * stride + nbase + col] = acc[r] * m;
  }
}

__device__ __forceinline__ v8f bias_acc(float bv) {
  v8f a = {bv, bv, bv, bv, bv, bv, bv, bv};
  return a;
}

// ---------------------------------------------------------------------------
// Kernel A: inverse flow. One wave (32 threads) per 16-sample tile.
// Stores zs[layer] = input of forward layer `layer`, writes log_prob.
// ---------------------------------------------------------------------------
__global__ __launch_bounds__(32)
void inv_flow_kernel(const float* __restrict__ y,
                     const float* __restrict__ W1, const float* __restrict__ b1,
                     const float* __restrict__ W2, const float* __restrict__ b2,
                     const float* __restrict__ W3, const float* __restrict__ b3,
                     float* __restrict__ zs, float* __restrict__ out_logprob) {
  __shared__ float zbuf[16 * D_INN];
  __shared__ float h1[16 * HID];
  __shared__ float h2[16 * HID];
  __shared__ float prmb[16 * 16];
  const int lane = threadIdx.x;
  const int tile = blockIdx.x;

  for (int e = lane; e < 16 * D_INN; e += 32) zbuf[e] = y[(size_t)tile * 192 + e];
  __syncthreads();

  float ld = 0.f;
  for (int layer = NLAYERS - 1; layer >= 0; --layer) {
    const float* W1l = W1 + (size_t)layer * HID * HALFD;
    const float* b1l = b1 + (size_t)layer * HID;
    const float* W2l = W2 + (size_t)layer * HID * HID;
    const float* b2l = b2 + (size_t)layer * HID;
    const float* W3l = W3 + (size_t)layer * D_INN * HID;
    const float* b3l = b3 + (size_t)layer * D_INN;

    // h1 = relu(z1 @ W1^T + b1); inverse step uses z1 = z[:,6:12]
    {
      v2f a0 = aT(zbuf + 6, D_INN, 0, HALFD);
      v2f a1 = aT(zbuf + 6, D_INN, 4, HALFD);
#pragma unroll
      for (int nt = 0; nt < HID / 16; ++nt) {
        int nb = nt * 16;
        v8f acc = bias_acc(b1l[nb + (lane & 15)]);
        acc = wmma4(a0, bT_k(W1l, HALFD, nb, 0, HALFD), acc);
        acc = wmma4(a1, bT_k(W1l, HALFD, nb, 4, HALFD), acc);
        store_relu(h1, HID, nb, acc);
      }
    }
    __syncthreads();
    // h2 = relu(h1 @ W2^T + b2): k-outer, 12 accumulator chains
    {
      v8f acc[HID / 16];
#pragma unroll
      for (int nt = 0; nt < HID / 16; ++nt)
        acc[nt] = bias_acc(b2l[nt * 16 + (lane & 15)]);
#pragma unroll 2
      for (int k0 = 0; k0 < HID; k0 += 4) {
        v2f a = aT_f(h1, HID, k0);
#pragma unroll
        for (int nt = 0; nt < HID / 16; ++nt)
          acc[nt] = wmma4(a, bT_f(W2l, HID, nt * 16, k0), acc[nt]);
      }
#pragma unroll
      for (int nt = 0; nt < HID / 16; ++nt) store_relu(h2, HID, nt * 16, acc[nt]);
    }
    __syncthreads();
    // prm = h2 @ W3^T + b3  (12 valid cols)
    {
      int col = lane & 15;
      v8f acc = bias_acc((col < D_INN) ? b3l[col] : 0.f);
#pragma unroll 8
      for (int k0 = 0; k0 < HID; k0 += 4)
        acc = wmma4(aT_f(h2, HID, k0), bT_n(W3l, HID, 0, k0, D_INN), acc);
      store_tile(prmb, 16, 0, acc);
    }
    __syncthreads();
    // coupling-inverse epilogue: lanes 0..15 own one sample each
    if (lane < 16) {
      const int i = lane;
      float z1v[HALFD], z2v[HALFD];
#pragma unroll
      for (int j = 0; j < HALFD; ++j) {
        z1v[j] = zbuf[i * D_INN + 6 + j];
        z2v[j] = zbuf[i * D_INN + j];
      }
      float ssum = 0.f;
#pragma unroll
      for (int j = 0; j < HALFD; ++j) {
        float shift = prmb[i * 16 + 2 * j];
        float sraw  = prmb[i * 16 + 2 * j + 1];
        float s = CLAMPC * tanhf(sraw * (1.0f / CLAMPC));
        ssum += s;
        z2v[j] = (z2v[j] - shift) * expf(-s);
      }
      ld -= ssum;
#pragma unroll
      for (int j = 0; j < HALFD; ++j) {
        zbuf[i * D_INN + j]     = z1v[j];
        zbuf[i * D_INN + 6 + j] = z2v[j];
      }
    }
    __syncthreads();
    // checkpoint forward-layer input
    float* zsl = zs + (size_t)layer * BATCH * D_INN + (size_t)tile * 192;
    for (int e = lane; e < 192; e += 32) zsl[e] = zbuf[e];
    __syncthreads();
  }

  if (lane < 16) {
    float sq = 0.f;
#pragma unroll
    for (int j = 0; j < D_INN; ++j) {
      float v = zbuf[lane * D_INN + j];
      sq += v * v;
    }
    // ld - 0.5*12*log(2pi) - 0.5*|z|^2
    out_logprob[(size_t)tile * 16 + lane] = ld - 11.0272626467149716f - 0.5f * sq;
  }
}

// ---------------------------------------------------------------------------
// Kernel B: VJP of flow_forward applied to 6 cotangent rows per sample.
// One wave per 16-sample tile; layers processed 23 -> 0.
// ---------------------------------------------------------------------------
__global__ __launch_bounds__(32)
void vjp_kernel(const float* __restrict__ Jp_in,
                const float* __restrict__ W1, const float* __restrict__ b1,
                const float* __restrict__ W2, const float* __restrict__ b2,
                const float* __restrict__ W3, const float* __restrict__ b3,
                const float* __restrict__ zs, float* __restrict__ jp_out) {
  __shared__ float zbuf[16 * D_INN];
  __shared__ float h1[16 * HID];
  __shared__ float h2[16 * HID];
  __shared__ float ga[16 * HID];
  __shared__ float gb[16 * HID];
  __shared__ float sbuf[16 * HALFD];   // s
  __shared__ float sebuf[16 * HALFD];  // exp(s)
  __shared__ float zebuf[16 * HALFD];  // z2*exp(s)
  __shared__ float dpm[16 * 16];       // prm / dprm / dz1 staging
  __shared__ float Ubuf[96 * D_INN];   // cotangents: row r = t*16 + sample
  const int lane = threadIdx.x;
  const int tile = blockIdx.x;

  // init cotangents from J_p
  for (int e = lane; e < 96 * D_INN; e += 32) {
    int r = e / D_INN, j = e % D_INN;
    int t = r >> 4, i = r & 15;
    Ubuf[e] = Jp_in[((size_t)(tile * 16 + i) * OUTD + t) * D_INN + j];
  }
  __syncthreads();

  for (int layer = NLAYERS - 1; layer >= 0; --layer) {
    const float* W1l = W1 + (size_t)layer * HID * HALFD;
    const float* b1l = b1 + (size_t)layer * HID;
    const float* W2l = W2 + (size_t)layer * HID * HID;
    const float* b2l = b2 + (size_t)layer * HID;
    const float* W3l = W3 + (size_t)layer * D_INN * HID;
    const float* b3l = b3 + (size_t)layer * D_INN;

    const float* zsl = zs + (size_t)layer * BATCH * D_INN + (size_t)tile * 192;
    for (int e = lane; e < 192; e += 32) zbuf[e] = zsl[e];
    __syncthreads();

    // ---- forward recompute (forward coupling: z1 = z[:,0:6]) ----
    {
      v2f a0 = aT(zbuf, D_INN, 0, HALFD);
      v2f a1 = aT(zbuf, D_INN, 4, HALFD);
#pragma unroll
      for (int nt = 0; nt < HID / 16; ++nt) {
        int nb = nt * 16;
        v8f acc = bias_acc(b1l[nb + (lane & 15)]);
        acc = wmma4(a0, bT_k(W1l, HALFD, nb, 0, HALFD), acc);
        acc = wmma4(a1, bT_k(W1l, HALFD, nb, 4, HALFD), acc);
        store_relu(h1, HID, nb, acc);
      }
    }
    __syncthreads();
    {
      v8f acc[HID / 16];
#pragma unroll
      for (int nt = 0; nt < HID / 16; ++nt)
        acc[nt] = bias_acc(b2l[nt * 16 + (lane & 15)]);
#pragma unroll 2
      for (int k0 = 0; k0 < HID; k0 += 4) {
        v2f a = aT_f(h1, HID, k0);
#pragma unroll
        for (int nt = 0; nt < HID / 16; ++nt)
          acc[nt] = wmma4(a, bT_f(W2l, HID, nt * 16, k0), acc[nt]);
      }
#pragma unroll
      for (int nt = 0; nt < HID / 16; ++nt) store_relu(h2, HID, nt * 16, acc[nt]);
    }
    __syncthreads();
    {
      int col = lane & 15;
      v8f acc = bias_acc((col < D_INN) ? b3l[col] : 0.f);
#pragma unroll 8
      for (int k0 = 0; k0 < HID; k0 += 4)
        acc = wmma4(aT_f(h2, HID, k0), bT_n(W3l, HID, 0, k0, D_INN), acc);
      store_tile(dpm, 16, 0, acc);  // prm
    }
    __syncthreads();
    if (lane < 16) {
      const int i = lane;
#pragma unroll
      for (int j = 0; j < HALFD; ++j) {
        float sraw = dpm[i * 16 + 2 * j + 1];
        float s = CLAMPC * tanhf(sraw * (1.0f / CLAMPC));
        float es = expf(s);
        sbuf[i * HALFD + j] = s;
        sebuf[i * HALFD + j] = es;
        zebuf[i * HALFD + j] = zbuf[i * D_INN + 6 + j] * es;  // z2*exp(s)
      }
    }
    __syncthreads();

    // ---- backward for each of the 6 tangents ----
    for (int t = 0; t < OUTD; ++t) {
      float ubr[HALFD];  // lanes 0..15 keep u_b in registers
      if (lane < 16) {
        const int i = lane;
        const int r = t * 16 + i;
#pragma unroll
        for (int j = 0; j < HALFD; ++j) {
          float ua = Ubuf[r * D_INN + j];
          float ub = Ubuf[r * D_INN + 6 + j];
          ubr[j] = ub;
          float s = sbuf[i * HALFD + j];
          float ds = ua * zebuf[i * HALFD + j];
          dpm[i * 16 + 2 * j]     = ua;                         // d shift
          dpm[i * 16 + 2 * j + 1] = ds * (1.f - 0.25f * s * s); // d sraw
          Ubuf[r * D_INN + 6 + j] = ua * sebuf[i * HALFD + j];  // new cot z2
        }
      } else {
#pragma unroll
        for (int j = 0; j < HALFD; ++j) ubr[j] = 0.f;
      }
      __syncthreads();
      // g2 = (dprm @ W3) * 1{h2>0}; K=12 is exactly 3 full k-steps
      {
        v2f a0 = aT_f(dpm, 16, 0);
        v2f a1 = aT_f(dpm, 16, 4);
        v2f a2 = aT_f(dpm, 16, 8);
#pragma unroll
        for (int nt = 0; nt < HID / 16; ++nt) {
          int nb = nt * 16;
          v8f acc = bias_acc(0.f);
          acc = wmma4(a0, bD_f(W3l, HID, nb, 0), acc);
          acc = wmma4(a1, bD_f(W3l, HID, nb, 4), acc);
          acc = wmma4(a2, bD_f(W3l, HID, nb, 8), acc);
          store_mask(ga, HID, nb, acc, h2, HID);
        }
      }
      __syncthreads();
      // g1 = (g2 @ W2) * 1{h1>0}: k-outer, 12 accumulator chains
      {
        v8f acc[HID / 16];
#pragma unroll
        for (int nt = 0; nt < HID / 16; ++nt) acc[nt] = bias_acc(0.f);
#pragma unroll 2
        for (int k0 = 0; k0 < HID; k0 += 4) {
          v2f a = aT_f(ga, HID, k0);
#pragma unroll
          for (int nt = 0; nt < HID / 16; ++nt)
            acc[nt] = wmma4(a, bD_f(W2l, HID, nt * 16, k0), acc[nt]);
        }
#pragma unroll
        for (int nt = 0; nt < HID / 16; ++nt)
          store_mask(gb, HID, nt * 16, acc[nt], h1, HID);
      }
      __syncthreads();
      // dz1 = g1 @ W1  (6 valid cols)
      {
        v8f acc = bias_acc(0.f);
#pragma unroll 8
        for (int k0 = 0; k0 < HID; k0 += 4)
          acc = wmma4(aT_f(gb, HID, k0), bD_n(W1l, HALFD, 0, k0, HALFD), acc);
        store_tile(dpm, 16, 0, acc);
      }
      __syncthreads();
      if (lane < 16) {
        const int i = lane;
        const int r = t * 16 + i;
#pragma unroll
        for (int j = 0; j < HALFD; ++j)
          Ubuf[r * D_INN + j] = ubr[j] + dpm[i * 16 + j];  // new cot z1
      }
      __syncthreads();
    }
  }

  // write pulled-back tangents: jp[b*72 + t*12 + j]
  for (int e = lane; e < 96 * D_INN; e += 32) {
    int r = e / D_INN, j = e % D_INN;
    int t = r >> 4, i = r & 15;
    jp_out[((size_t)(tile * 16 + i) * OUTD + t) * D_INN + j] = Ubuf[e];
  }
}

// ---------------------------------------------------------------------------
// Deterministic two-stage reduction of sum(Jp^2) -> red[256] = 1/sqrt(sum)
// ---------------------------------------------------------------------------
__global__ __launch_bounds__(256)
void reduce1_kernel(const float* __restrict__ jp, float* __restrict__ red) {
  __shared__ float sm[256];
  float acc = 0.f;
  const int total = BATCH * OUTD * D_INN;
  for (int idx = blockIdx.x * 256 + threadIdx.x; idx < total; idx += 256 * 256) {
    float v = jp[idx];
    acc += v * v;
  }
  sm[threadIdx.x] = acc;
  __syncthreads();
  for (int s = 128; s > 0; s >>= 1) {
    if ((int)threadIdx.x < s) sm[threadIdx.x] += sm[threadIdx.x + s];
    __syncthreads();
  }
  if (threadIdx.x == 0) red[blockIdx.x] = sm[0];
}

__global__ __launch_bounds__(256)
void reduce2_kernel(float* __restrict__ red) {
  __shared__ float sm[256];
  sm[threadIdx.x] = red[threadIdx.x];
  __syncthreads();
  for (int s = 128; s > 0; s >>= 1) {
    if ((int)threadIdx.x < s) sm[threadIdx.x] += sm[threadIdx.x + s];
    __syncthreads();
  }
  if (threadIdx.x == 0) red[256] = 1.0f / sqrtf(sm[0]);
}

// ---------------------------------------------------------------------------
// Kernel D: per-sample KL (packed lower-tri Cholesky).
// ---------------------------------------------------------------------------
__device__ __forceinline__ float chol_logdet(float* A, int N) {
  float ld = 0.f;
  for (int n = 0; n < N; ++n) {
    for (int m = 0; m <= n; ++m) {
      float s = A[n * (n + 1) / 2 + m];
      for (int k = 0; k < m; ++k)
        s -= A[n * (n + 1) / 2 + k] * A[m * (m + 1) / 2 + k];
      if (m == n) {
        float d = sqrtf(s);
        A[n * (n + 1) / 2 + n] = d;
        ld += logf(d);
      } else {
        A[n * (n + 1) / 2 + m] = s / A[m * (m + 1) / 2 + m];
      }
    }
  }
  return 2.f * ld;
}

__global__ __launch_bounds__(64)
void kl_kernel(const float* __restrict__ zs, const float* __restrict__ jp,
               const float* __restrict__ red, const float* __restrict__ Wsym,
               const float* __restrict__ lvd, float* __restrict__ out) {
  int b = blockIdx.x * 64 + threadIdx.x;
  if (b >= BATCH) return;
  const float scale = red[256];

  float z[D_INN];
#pragma unroll
  for (int j = 0; j < D_INN; ++j) z[j] = zs[(size_t)b * D_INN + j];
  float Jp[OUTD][D_INN];
#pragma unroll
  for (int n = 0; n < OUTD; ++n)
#pragma unroll
    for (int j = 0; j < D_INN; ++j)
      Jp[n][j] = jp[(size_t)b * 72 + n * D_INN + j] * scale;

  // M = Jp Jp^T + EPS*I  (packed lower 6x6)
  float M[21];
  {
    int p = 0;
    for (int n = 0; n < OUTD; ++n)
      for (int m = 0; m <= n; ++m) {
        float s = 0.f;
#pragma unroll
        for (int i = 0; i < D_INN; ++i) s += Jp[n][i] * Jp[m][i];
        if (m == n) s += EPSP;
        M[p++] = s;
      }
  }
  float dm = 0.f;
  for (int n = 0; n < OUTD; ++n) dm += M[n * (n + 1) / 2 + n];
  float norm_M = fmaxf(dm / OUTD, 1e-6f);
  for (int n = 0; n < OUTD; ++n) M[n * (n + 1) / 2 + n] += 0.001f * norm_M;

  // accumulate over 66 antisymmetric bases
  float Sq[78];
  for (int p = 0; p < 78; ++p) Sq[p] = 0.f;
  float trq = 0.f, trpq = 0.f;
  for (int ib = 0; ib < NBASES; ++ib) {
    const float* W = Wsym + (size_t)ib * 144;
    float Qi[D_INN];
#pragma unroll
    for (int j = 0; j < D_INN; ++j) {
      float a = 0.f;
#pragma unroll
      for (int k = 0; k < D_INN; ++k) a += z[k] * (W[j * 12 + k] - W[k * 12 + j]);
      Qi[j] = a;
      trq += a * a;
    }
    int p = 0;
    for (int n = 0; n < D_INN; ++n)
      for (int m = 0; m <= n; ++m) Sq[p++] += Qi[n] * Qi[m];
#pragma unroll
    for (int n = 0; n < OUTD; ++n) {
      float s = 0.f;
#pragma unroll
      for (int j = 0; j < D_INN; ++j) s += Jp[n][j] * Qi[j];
      trpq += s * s;
    }
  }

  float D_[D_INN], Db[D_INN];
#pragma unroll
  for (int j = 0; j < D_INN; ++j) D_[j] = expf(-lvd[j]);
  float dh = 0.f;
  for (int n = 0; n < D_INN; ++n) dh += Sq[n * (n + 1) / 2 + n] + D_[n];
  float norm_H = fmaxf(dh / D_INN, 1e-6f);
  float sumDb = 0.f;
#pragma unroll
  for (int j = 0; j < D_INN; ++j) {
    Db[j] = D_[j] + 0.001f * norm_H;
    sumDb += Db[j];
  }
  for (int n = 0; n < D_INN; ++n)
    Sq[n * (n + 1) / 2 + n] += D_[n] + 0.001f * norm_H;  // H

  float trp = 0.f;
#pragma unroll
  for (int n = 0; n < OUTD; ++n)
#pragma unroll
    for (int j = 0; j < D_INN; ++j) trp += Jp[n][j] * Jp[n][j] * Db[j];

  float trace = (EPSP + 0.001f * norm_M) * (sumDb + trq) + trp + trpq;
  float logdet_M = chol_logdet(M, OUTD);
  float logdet_H = chol_logdet(Sq, D_INN);
  float logdet_p = logdet_M + (float)(D_INN - OUTD) * logf(EPSP);
  float logdet_q = logdet_H;  // Db[:, SYM:] is empty (SYM == D_IN)
  float logdet = -(logdet_p + logdet_q);
  out[BATCH + b] = 0.5f * (trace + logdet - (float)D_INN);
}

// ---------------------------------------------------------------------------
extern "C" void kernel_launch(void* const* d_in, const int* in_sizes, int n_in,
                              void* d_out, int out_size, void* d_ws, size_t ws_size,
                              hipStream_t stream) {
  (void)in_sizes; (void)n_in; (void)out_size; (void)ws_size;
  const float* y    = (const float*)d_in[0];
  const float* Jp   = (const float*)d_in[1];
  const float* W1   = (const float*)d_in[2];
  const float* b1   = (const float*)d_in[3];
  const float* W2   = (const float*)d_in[4];
  const float* b2   = (const float*)d_in[5];
  const float* W3   = (const float*)d_in[6];
  const float* b3   = (const float*)d_in[7];
  const float* Wsym = (const float*)d_in[8];
  const float* lvd  = (const float*)d_in[9];
  float* out = (float*)d_out;

  float* ws  = (float*)d_ws;
  float* zs  = ws;                                      // 24*B*12 floats
  float* jp  = ws + (size_t)NLAYERS * BATCH * D_INN;    // B*72 floats
  float* red = jp + (size_t)BATCH * OUTD * D_INN;       // 257 floats

  const int tiles = BATCH / 16;
  hipLaunchKernelGGL(inv_flow_kernel, dim3(tiles), dim3(32), 0, stream,
                     y, W1, b1, W2, b2, W3, b3, zs, out);
  hipLaunchKernelGGL(vjp_kernel, dim3(tiles), dim3(32), 0, stream,
                     Jp, W1, b1, W2, b2, W3, b3, zs, jp);
  hipLaunchKernelGGL(reduce1_kernel, dim3(256), dim3(256), 0, stream, jp, red);
  hipLaunchKernelGGL(reduce2_kernel, dim3(1), dim3(256), 0, stream, red);
  hipLaunchKernelGGL(kl_kernel, dim3((BATCH + 63) / 64), dim3(64), 0, stream,
                     zs, jp, red, Wsym, lvd, out);
}